// STGRIT_18073222381656
// MI455X (gfx1250) — compile-verified
//
#include <hip/hip_runtime.h>
#include <hip/hip_bf16.h>

// ---------------------------------------------------------------------------
// GRIT sparse graph attention, fp32, WMMA f32 16x16x4 for the two real GEMMs.
// Attention logit algebraically reduced to matvecs (see analysis).
// ---------------------------------------------------------------------------

typedef float v2f __attribute__((ext_vector_type(2)));
typedef float v8f __attribute__((ext_vector_type(8)));

#define D_MODEL 128
#define D_EDGE  32
#define NNODE   10000
#define RSQRT_DH 0.17677669529663689f  // 1/sqrt(32)

// workspace layout (float offsets)
#define WS_V      0            // [40000*128] V = x@Wv + bv
#define WS_QA     5120000      // [40000]     qa = x.wqa + bq.Wa
#define WS_KA     5160000      // [40000]     ka
#define WS_ASUM   5200000      // [40000]     segment softmax denominators
#define WS_AEXP   5240000      // [640000]    exp(logit) per edge
#define WS_WQA    5880000      // [128]       Wq @ Wa
#define WS_WKA    5880128      // [128]       Wk @ Wa
#define WS_WEWA   5880256      // [32]        Wew @ Wa
#define WS_CONST  5880288      // [3] bq.Wa, bk.Wa, bew.Wa
// total ~23.5 MB of scratch

// ---- precompute folded weight vectors (1 block, 128 threads) --------------
__global__ void grit_pre(const float* __restrict__ Wq, const float* __restrict__ Wk,
                         const float* __restrict__ Wew,
                         const float* __restrict__ bq, const float* __restrict__ bk,
                         const float* __restrict__ bew,
                         const float* __restrict__ Wa, float* __restrict__ ws) {
  const int t = threadIdx.x;  // 0..127
  float aq = 0.f, ak = 0.f;
  for (int j = 0; j < D_MODEL; ++j) {
    const float w = Wa[j];
    aq += Wq[t * D_MODEL + j] * w;
    ak += Wk[t * D_MODEL + j] * w;
  }
  ws[WS_WQA + t] = aq;
  ws[WS_WKA + t] = ak;
  if (t < D_EDGE) {
    float ae = 0.f;
    for (int j = 0; j < D_MODEL; ++j) ae += Wew[t * D_MODEL + j] * Wa[j];
    ws[WS_WEWA + t] = ae;
  }
  if (t == 0) {
    float cq = 0.f, ck = 0.f, ce = 0.f;
    for (int j = 0; j < D_MODEL; ++j) {
      const float w = Wa[j];
      cq += bq[j] * w; ck += bk[j] * w; ce += bew[j] * w;
    }
    ws[WS_CONST + 0] = cq; ws[WS_CONST + 1] = ck; ws[WS_CONST + 2] = ce;
  }
}

// ---- node phase: V = x@Wv + bv via WMMA; qa/ka matvecs --------------------
// block = 256 threads (8 waves); block handles 16 rows x 128 cols.
__global__ void grit_nodes(const float* __restrict__ x, const float* __restrict__ Wv,
                           const float* __restrict__ bv, float* __restrict__ ws) {
  const int row0 = blockIdx.x * 16;
  const int lane = threadIdx.x & 31;
  const int wv   = threadIdx.x >> 5;       // wave id 0..7 -> column tile
  const int col0 = wv * 16;
  const int m    = lane & 15;              // row/col within tile (ISA frag layout)
  const int kb   = (lane >> 4) * 2;        // K sub-offset for A/B frags

  // accumulator pre-loaded with bias (every row of D gets +bv[col])
  v8f c;
  {
    const float bias = bv[col0 + m];
    for (int r = 0; r < 8; ++r) c[r] = bias;
  }

  const float* xrow = x + (size_t)(row0 + m) * D_MODEL;
  for (int k0 = 0; k0 < D_MODEL; k0 += 4) {
    v2f a, b;
    a.x = xrow[k0 + kb];
    a.y = xrow[k0 + kb + 1];
    b.x = Wv[(k0 + kb) * D_MODEL + col0 + m];
    b.y = Wv[(k0 + kb + 1) * D_MODEL + col0 + m];
    c = __builtin_amdgcn_wmma_f32_16x16x4_f32(false, a, false, b, (short)0, c,
                                              false, false);
  }

  float* V = ws + WS_V;
  const int rhi = 8 * (lane >> 4);
  for (int r = 0; r < 8; ++r)
    V[(size_t)(row0 + r + rhi) * D_MODEL + col0 + m] = c[r];

  // qa/ka: one thread per row, 128-length dots against folded vectors
  if (threadIdx.x < 16) {
    const int row = row0 + threadIdx.x;
    const float* xr  = x + (size_t)row * D_MODEL;
    const float* wqa = ws + WS_WQA;
    const float* wka = ws + WS_WKA;
    float aq = ws[WS_CONST + 0], ak = ws[WS_CONST + 1];
    for (int i = 0; i < D_MODEL; ++i) {
      const float xv = xr[i];
      aq += xv * wqa[i];
      ak += xv * wka[i];
    }
    ws[WS_QA + row] = aq;
    ws[WS_KA + row] = ak;
  }
}

// ---- edge logits: exp + segment-sum ---------------------------------------
__global__ void grit_logits(const int* __restrict__ ei, const float* __restrict__ e_emb,
                            const int* __restrict__ bidx, float* __restrict__ ws,
                            int E) {
  const int e = blockIdx.x * blockDim.x + threadIdx.x;
  if (e >= E) return;
  const int s = ei[e], d = ei[E + e], b = bidx[e];
  const float4* er = (const float4*)(e_emb + (size_t)e * D_EDGE);
  const float4* wr = (const float4*)(ws + WS_WEWA);
  float dotv = ws[WS_CONST + 2];
  for (int i = 0; i < D_EDGE / 4; ++i) {
    const float4 ev = er[i], wv = wr[i];
    dotv += ev.x * wv.x + ev.y * wv.y + ev.z * wv.z + ev.w * wv.w;
  }
  const int g = b * NNODE + d;
  const float logit = (ws[WS_QA + b * NNODE + s] + ws[WS_KA + g] + dotv) * RSQRT_DH;
  const float p = expf(logit);
  ws[WS_AEXP + e] = p;
  __hip_atomic_fetch_add(ws + WS_ASUM + g, p, __ATOMIC_RELAXED,
                         __HIP_MEMORY_SCOPE_AGENT);
}

// ---- edge messages: Ev via WMMA (in-register), fused scatter --------------
// block = 256 threads (8 waves) handles 16 edges; wave w does cols [16w,16w+16).
__global__ void grit_msg(const int* __restrict__ ei, const float* __restrict__ e_emb,
                         const int* __restrict__ bidx, const float* __restrict__ Wev,
                         const float* __restrict__ bev, const float* __restrict__ ws,
                         float* __restrict__ out, int E) {
  __shared__ float s_alpha[16];
  __shared__ int   s_vbase[16];
  __shared__ int   s_obase[16];

  const int e0 = blockIdx.x * 16;
  if (threadIdx.x < 16) {
    const int e = e0 + threadIdx.x;
    const int s = ei[e], d = ei[E + e], b = bidx[e];
    const int g = b * NNODE + d;
    s_alpha[threadIdx.x] = ws[WS_AEXP + e] / (ws[WS_ASUM + g] + 1e-9f);
    s_vbase[threadIdx.x] = g * D_MODEL;
    s_obase[threadIdx.x] = (b * NNODE + s) * D_MODEL;
  }
  __syncthreads();

  const int lane = threadIdx.x & 31;
  const int wv   = threadIdx.x >> 5;
  const int col0 = wv * 16;
  const int m    = lane & 15;
  const int kb   = (lane >> 4) * 2;

  v8f c;
  {
    const float bias = bev[col0 + m];
    for (int r = 0; r < 8; ++r) c[r] = bias;   // Ev + bev folded into acc init
  }

  const float* er = e_emb + (size_t)(e0 + m) * D_EDGE;
  for (int k0 = 0; k0 < D_EDGE; k0 += 4) {
    v2f a, b;
    a.x = er[k0 + kb];
    a.y = er[k0 + kb + 1];
    b.x = Wev[(k0 + kb) * D_MODEL + col0 + m];
    b.y = Wev[(k0 + kb + 1) * D_MODEL + col0 + m];
    c = __builtin_amdgcn_wmma_f32_16x16x4_f32(false, a, false, b, (short)0, c,
                                              false, false);
  }

  const float* V = ws + WS_V;
  const int rhi = 8 * (lane >> 4);
  const int col = col0 + m;
  for (int r = 0; r < 8; ++r) {
    const int row = r + rhi;
    const float msg = s_alpha[row] * (V[s_vbase[row] + col] + c[r]);
    __hip_atomic_fetch_add(out + s_obase[row] + col, msg, __ATOMIC_RELAXED,
                           __HIP_MEMORY_SCOPE_AGENT);
  }
}

// ---------------------------------------------------------------------------
extern "C" void kernel_launch(void* const* d_in, const int* in_sizes, int n_in,
                              void* d_out, int out_size, void* d_ws, size_t ws_size,
                              hipStream_t stream) {
  const float* x    = (const float*)d_in[0];
  const int*   ei   = (const int*)  d_in[1];
  const float* eemb = (const float*)d_in[2];
  const int*   bi   = (const int*)  d_in[3];
  const float* Wq   = (const float*)d_in[4];
  const float* bq   = (const float*)d_in[5];
  const float* Wk   = (const float*)d_in[6];
  const float* bk   = (const float*)d_in[7];
  const float* Wv   = (const float*)d_in[8];
  const float* bv   = (const float*)d_in[9];
  const float* Wew  = (const float*)d_in[10];
  const float* bew  = (const float*)d_in[11];
  const float* Wev  = (const float*)d_in[12];
  const float* bev  = (const float*)d_in[13];
  const float* Wa   = (const float*)d_in[14];

  float* ws  = (float*)d_ws;
  float* out = (float*)d_out;
  const int E  = in_sizes[3];              // 640000
  const int BN = in_sizes[0] / D_MODEL;    // 40000

  hipMemsetAsync(ws + WS_ASUM, 0, (size_t)BN * sizeof(float), stream);
  hipMemsetAsync(out, 0, (size_t)out_size * sizeof(float), stream);

  grit_pre   <<<1,              128, 0, stream>>>(Wq, Wk, Wew, bq, bk, bew, Wa, ws);
  grit_nodes <<<BN / 16,        256, 0, stream>>>(x, Wv, bv, ws);
  grit_logits<<<(E + 255)/256,  256, 0, stream>>>(ei, eemb, bi, ws, E);
  grit_msg   <<<E / 16,         256, 0, stream>>>(ei, eemb, bi, Wev, bev, ws, out, E);
}